// VoteFusion_22952305229833
// MI455X (gfx1250) — compile-verified
//
#include <hip/hip_runtime.h>
#include <math.h>

// VoteFusion fused kernel for gfx1250 (MI455X).
// Distance argmin over boxes done with V_WMMA_F32_16X16X4_F32:
//   score[m][n] = -2*mx_m*sx_n - 2*my_m*sy_n + (mx_m^2+my_m^2)
//              = du^2+dv^2 - (sx^2+sy^2)   (seed-constant offset, argmin-safe)

typedef __attribute__((ext_vector_type(2))) float v2f;
typedef __attribute__((ext_vector_type(8))) float v8f;

#define NBOX 512
#define HH 480
#define WW 640
#define C_OUT 18
#define SEEDS_PER_BLOCK 128

__global__ __launch_bounds__(256) void vote_fusion_kernel(
    const float* __restrict__ img,    // (B,3,H,W)
    const float* __restrict__ bbox,   // (B,N,6) l,t,r,b,conf,cls
    const float* __restrict__ seeds,  // (B,S,3)
    const float* __restrict__ Rt,     // (B,3,3)
    const float* __restrict__ Km,     // (B,3,3)
    float* __restrict__ out,          // (B,18,S)
    int Sn)
{
    __shared__ float  sA0[NBOX];    // -2*midx
    __shared__ float  sA1[NBOX];    // -2*midy
    __shared__ float  sA2[NBOX];    // midx^2+midy^2
    __shared__ float4 sBnd[NBOX];   // l,t,r,b
    __shared__ float  sMx[NBOX], sMy[NBOX], sCf[NBOX];
    __shared__ int    sCl[NBOX];

    const int blocksPerBatch = Sn / SEEDS_PER_BLOCK;
    const int b  = blockIdx.x / blocksPerBatch;
    const int s0 = (blockIdx.x % blocksPerBatch) * SEEDS_PER_BLOCK;

    // ---- stage boxes into LDS (uniform: 2 boxes per thread) ----
    for (int j = threadIdx.x; j < NBOX; j += blockDim.x) {
        const float* bb = bbox + ((size_t)b * NBOX + j) * 6;
        float l = bb[0], t = bb[1], r = bb[2], btm = bb[3];
        float mx = (l + r) * 0.5f, my = (t + btm) * 0.5f;
        sA0[j] = -2.0f * mx;
        sA1[j] = -2.0f * my;
        sA2[j] = mx * mx + my * my;
        sBnd[j] = make_float4(l, t, r, btm);
        sMx[j] = mx; sMy[j] = my; sCf[j] = bb[4]; sCl[j] = (int)bb[5];
    }
    __syncthreads();

    const int wave  = threadIdx.x >> 5;
    const int lane  = threadIdx.x & 31;
    const int nIdx  = lane & 15;            // seed slot within wave's 16-seed tile
    const bool hiH  = lane >= 16;
    const int s     = s0 + wave * 16 + nIdx;

    // ---- camera / rotation (uniform per block) ----
    const float* R  = Rt + (size_t)b * 9;
    const float* Kp = Km + (size_t)b * 9;
    const float fu = Kp[0], cx0 = Kp[2], fv = Kp[4], cy0 = Kp[5];
    const float r00=R[0], r01=R[1], r02=R[2],
                r10=R[3], r11=R[4], r12=R[5],
                r20=R[6], r21=R[7], r22=R[8];

    // ---- project seed (both half-waves compute the same 16 seeds) ----
    const float* sp = seeds + ((size_t)b * Sn + s) * 3;
    const float p0 = sp[0], p1 = sp[1], p2 = sp[2];
    // q = Rt * p ; cam = [q0, -q2, q1]
    const float q0 = r00*p0 + r01*p1 + r02*p2;
    const float q1 = r10*p0 + r11*p1 + r12*p2;
    const float q2 = r20*p0 + r21*p1 + r22*p2;
    const float camx = q0, camy = -q2, camz = q1;
    const float sx = (fu * camx + cx0 * camz) / camz;   // unclipped seed_2d
    const float sy = (fv * camy + cy0 * camz) / camz;

    // ---- B fragment (4x16, K x N): lanes 0-15 hold K0,K1; 16-31 hold K2,K3 ----
    v2f bfrag;
    bfrag[0] = hiH ? 1.0f : sx;
    bfrag[1] = hiH ? 0.0f : sy;

    const v8f czero = {0.f,0.f,0.f,0.f,0.f,0.f,0.f,0.f};
    float best  = 3.402823e38f;
    int   besti = 0;
    const int mOff = hiH ? 8 : 0;

    // ---- 32 box tiles of 16: WMMA + lane-local running argmin ----
    for (int t = 0; t < NBOX / 16; ++t) {
        const int am = t * 16 + nIdx;
        v2f afrag;  // A 16x4: lanes 0-15 -> (K0,K1); lanes 16-31 -> (K2,K3)
        if (!hiH) { afrag[0] = sA0[am]; afrag[1] = sA1[am]; }
        else      { afrag[0] = sA2[am]; afrag[1] = 0.0f;    }
        v8f d = __builtin_amdgcn_wmma_f32_16x16x4_f32(
            false, afrag, false, bfrag, (short)0, czero, false, false);
#pragma unroll
        for (int rr = 0; rr < 8; ++rr) {
            const int m = t * 16 + mOff + rr;   // strictly increasing per lane
            const float dm = d[rr];
            if (dm < best) { best = dm; besti = m; }   // strict < => first min
        }
    }
    // merge half-lanes (same seed); tie -> smaller box index (jnp.argmin order)
    {
        const float ov = __shfl_xor(best, 16, 32);
        const int   oi = __shfl_xor(besti, 16, 32);
        if (ov < best || (ov == best && oi < besti)) { best = ov; besti = oi; }
    }

    // ---- in_box OR-scan: 256 boxes per half-lane off LDS ----
    int any = 0;
    const int jStart = hiH ? NBOX / 2 : 0;
    for (int j = jStart; j < jStart + NBOX / 2; ++j) {
        const float4 bd = sBnd[j];
        any |= (sx > bd.x) & (sx < bd.z) & (sy > bd.y) & (sy < bd.w);
    }
    any |= __shfl_xor(any, 16, 32);

    // ---- epilogue: lanes 0-15 write 18 channels for their seed ----
    if (!hiH) {
        // texture sample (clipped coords)
        const float sxc = fminf(fmaxf(sx, 0.0f), (float)(WW - 1));
        const float syc = fminf(fmaxf(sy, 0.0f), (float)(HH - 1));
        const int idx = (int)floorf(syc) * WW + (int)floorf(sxc);
        const float* imgB = img + (size_t)b * 3 * HH * WW;
        const float t0 = (imgB[idx]              + 103.530f - 128.0f) * (1.0f/128.0f);
        const float t1 = (imgB[HH*WW + idx]      + 116.280f - 128.0f) * (1.0f/128.0f);
        const float t2 = (imgB[2*HH*WW + idx]    + 123.675f - 128.0f) * (1.0f/128.0f);

        const int j = besti;
        const float mx = sMx[j], my = sMy[j];
        const float du = mx - sx, dv = my - sy;
        const float g0 = du * (camz / fu);
        const float g1 = dv * (camz / fv);
        // geo_xy = [g0,0,-g1] @ Rt, scaled by depth ratio
        const float gx0 = g0 * r00 - g1 * r20;
        const float gx1 = g0 * r01 - g1 * r21;
        const float gx2 = g0 * r02 - g1 * r22;
        const float ratio = p1 / (p1 + gx1);
        const float geoA = gx0 * ratio;
        const float geoB = gx2 * ratio;
        // geo_vec = [g0+camx, camz, -(g1+camy)] @ Rt, normalized
        const float a0 = g0 + camx, a1 = camz, a2 = -(g1 + camy);
        float gv0 = a0 * r00 + a1 * r10 + a2 * r20;
        float gv1 = a0 * r01 + a1 * r11 + a2 * r21;
        float gv2 = a0 * r02 + a1 * r12 + a2 * r22;
        const float inv = 1.0f / sqrtf(gv0*gv0 + gv1*gv1 + gv2*gv2);
        gv0 *= inv; gv1 *= inv; gv2 *= inv;

        const float conf = sCf[j];
        const int   cls  = sCl[j];
        const bool  valid = (any != 0);

        float pos[C_OUT];
#pragma unroll
        for (int c = 0; c < 10; ++c) pos[c] = (c == cls) ? conf : 0.0f;
        pos[10] = t0; pos[11] = t1; pos[12] = t2;
        pos[13] = geoA; pos[14] = geoB;
        pos[15] = gv0; pos[16] = gv1; pos[17] = gv2;

        float* outB = out + (size_t)b * C_OUT * Sn + s;
#pragma unroll
        for (int c = 0; c < C_OUT; ++c)
            outB[(size_t)c * Sn] = valid ? pos[c] : 0.0f;
    }
}

extern "C" void kernel_launch(void* const* d_in, const int* in_sizes, int n_in,
                              void* d_out, int out_size, void* d_ws, size_t ws_size,
                              hipStream_t stream) {
    const float* img   = (const float*)d_in[0];
    const float* bbox  = (const float*)d_in[1];
    const float* seeds = (const float*)d_in[2];
    const float* Rt    = (const float*)d_in[3];
    const float* Km    = (const float*)d_in[4];
    float* out = (float*)d_out;

    const int B = in_sizes[3] / 9;            // Rt is (B,3,3)
    const int S = in_sizes[2] / (3 * B);      // seeds is (B,S,3)

    dim3 grid(B * (S / SEEDS_PER_BLOCK));
    dim3 block(256);
    hipLaunchKernelGGL(vote_fusion_kernel, grid, block, 0, stream,
                       img, bbox, seeds, Rt, Km, out, S);
}